// SlotAttention_27058293965251
// MI455X (gfx1250) — compile-verified
//
#include <hip/hip_runtime.h>
#include <math.h>

// ---------------------------------------------------------------------------
// Slot Attention for MI455X (gfx1250). All matmuls use V_WMMA_F32_16X16X4_F32
// (full fp32 precision; workload is memory-bound: ~1.1 GB HBM traffic after
// fusing LN into the k/v projection, ~50us floor at 23.3 TB/s).
// ---------------------------------------------------------------------------

typedef __attribute__((ext_vector_type(2))) float v2f;
typedef __attribute__((ext_vector_type(8))) float v8f;

#define B_    64
#define N_    4096
#define D_    128
#define K_    8
#define MLPH_ 256

__device__ __forceinline__ v2f ld2(const float* p) { return *(const v2f*)p; }

__device__ __forceinline__ v8f wmma4(v2f a, v2f b, v8f c) {
  // D = A(16x4, f32) * B(4x16, f32) + C(16x16, f32)
  return __builtin_amdgcn_wmma_f32_16x16x4_f32(false, a, false, b, (short)0, c,
                                               false, false);
}

// ---- Fused: x = LN(inputs[16 rows]); k = x Wk^T; v = x Wv^T ----------------
// Block = 256 threads (8 waves) per 16-row tile. Normalized x staged in LDS;
// wave w computes output column tile w (16 cols) for BOTH k and v, sharing the
// A fragments (ds_load_b64) across the two WMMA accumulator chains.
__global__ __launch_bounds__(256)
void ln_kv_fused(const float* __restrict__ in, const float* __restrict__ g,
                 const float* __restrict__ b, const float* __restrict__ Wk,
                 const float* __restrict__ Wv, float* __restrict__ kout,
                 float* __restrict__ vout) {
  __shared__ float lds_x[16 * D_];  // 8 KB
  const int w = threadIdx.x >> 5;
  const int lane = threadIdx.x & 31;
  const size_t row0 = (size_t)blockIdx.x * 16;

  // --- LayerNorm: each wave normalizes 2 rows (wave-per-row reductions) ---
  float4 gg = ((const float4*)g)[lane];
  float4 bb = ((const float4*)b)[lane];
#pragma unroll
  for (int rr = 0; rr < 2; ++rr) {
    const int r = w * 2 + rr;
    float4 x = ((const float4*)(in + (row0 + r) * D_))[lane];
    float s = x.x + x.y + x.z + x.w;
    for (int m = 16; m; m >>= 1) s += __shfl_xor(s, m, 32);
    float mean = s * (1.0f / 128.0f);
    float dx = x.x - mean, dy = x.y - mean, dz = x.z - mean, dw = x.w - mean;
    float vs = dx * dx + dy * dy + dz * dz + dw * dw;
    for (int m = 16; m; m >>= 1) vs += __shfl_xor(vs, m, 32);
    float rstd = rsqrtf(vs * (1.0f / 128.0f) + 1e-5f);
    float4 o;
    o.x = dx * rstd * gg.x + bb.x;
    o.y = dy * rstd * gg.y + bb.y;
    o.z = dz * rstd * gg.z + bb.z;
    o.w = dw * rstd * gg.w + bb.w;
    ((float4*)(lds_x + r * D_))[lane] = o;
  }
  __syncthreads();

  // --- WMMA: col tile = wave id; A frags from LDS, W frags from L2 ---
  const int m = lane & 15, half = lane >> 4;
  const float* arow = lds_x + m * D_ + half * 2;
  const float* wkrow = Wk + (size_t)(w * 16 + m) * D_ + half * 2;
  const float* wvrow = Wv + (size_t)(w * 16 + m) * D_ + half * 2;
  v8f acck = {}, accv = {};
#pragma unroll
  for (int kb = 0; kb < D_; kb += 4) {
    v2f a = ld2(arow + kb);
    acck = wmma4(a, ld2(wkrow + kb), acck);
    accv = wmma4(a, ld2(wvrow + kb), accv);
  }
  const int col = w * 16 + m;
#pragma unroll
  for (int j = 0; j < 8; ++j) {
    const size_t row = row0 + j + half * 8;  // D layout: vgpr j -> M=j(+8)
    kout[row * D_ + col] = acck[j];
    vout[row * D_ + col] = accv[j];
  }
}

// ---------------- LayerNorm over rows of 128 (wave per row) -----------------
__global__ __launch_bounds__(256)
void ln_rows_128(const float* __restrict__ in, const float* __restrict__ g,
                 const float* __restrict__ b, float* __restrict__ out, int R) {
  int row = blockIdx.x * 8 + (threadIdx.x >> 5);
  if (row >= R) return;
  int lane = threadIdx.x & 31;
  float4 x = ((const float4*)(in + (size_t)row * D_))[lane];
  float s = x.x + x.y + x.z + x.w;
  for (int m = 16; m; m >>= 1) s += __shfl_xor(s, m, 32);
  float mean = s * (1.0f / 128.0f);
  float dx = x.x - mean, dy = x.y - mean, dz = x.z - mean, dw = x.w - mean;
  float vs = dx * dx + dy * dy + dz * dz + dw * dw;
  for (int m = 16; m; m >>= 1) vs += __shfl_xor(vs, m, 32);
  float rstd = rsqrtf(vs * (1.0f / 128.0f) + 1e-5f);
  float4 gg = ((const float4*)g)[lane];
  float4 bb = ((const float4*)b)[lane];
  float4 o;
  o.x = dx * rstd * gg.x + bb.x;
  o.y = dy * rstd * gg.y + bb.y;
  o.z = dz * rstd * gg.z + bb.z;
  o.w = dw * rstd * gg.w + bb.w;
  ((float4*)(out + (size_t)row * D_))[lane] = o;
}

// -------- Generic WMMA GEMM: out[MxNout] = A[MxKd] @ W[Nout x Kd]^T ---------
// grid = (M/16, Nout/16), one wave (32 threads) per 16x16 tile.
// ep: 0 = none, 1 = exact GELU. resid (optional) added after epilogue.
__global__ __launch_bounds__(32)
void gemm_xwt(const float* __restrict__ A, const float* __restrict__ W,
              const float* __restrict__ bias, const float* __restrict__ resid,
              float* __restrict__ out, int Kd, int Nout, int ep) {
  const int lane = threadIdx.x;
  const int m = lane & 15, half = lane >> 4;
  const int mt = blockIdx.x, nt = blockIdx.y;
  const float* arow = A + (size_t)(mt * 16 + m) * Kd + half * 2;
  const float* wrow = W + (size_t)(nt * 16 + m) * Kd + half * 2;
  v8f acc = {};
  for (int kb = 0; kb < Kd; kb += 4)
    acc = wmma4(ld2(arow + kb), ld2(wrow + kb), acc);
  const int col = nt * 16 + m;
  const float bia = bias ? bias[col] : 0.0f;
#pragma unroll
  for (int j = 0; j < 8; ++j) {
    const int row = mt * 16 + j + half * 8;
    float x = acc[j] + bia;
    if (ep == 1) x = 0.5f * x * (1.0f + erff(x * 0.70710678118654752f));
    if (resid) x += resid[(size_t)row * Nout + col];
    out[(size_t)row * Nout + col] = x;
  }
}

// -------- dots[b,k,n] = scale * q[b,k,:] . kbuf[b,n,:]  (M=8 padded to 16) --
__global__ __launch_bounds__(32)
void dots_qk(const float* __restrict__ q, const float* __restrict__ kbuf,
             float* __restrict__ dots) {
  const int lane = threadIdx.x;
  const int m = lane & 15, half = lane >> 4;
  const int nt = blockIdx.x, b = blockIdx.y;
  const float* qrow = q + (size_t)b * K_ * D_ + (size_t)m * D_ + half * 2;
  const float* krow =
      kbuf + (size_t)b * N_ * D_ + (size_t)(nt * 16 + m) * D_ + half * 2;
  v8f acc = {};
#pragma unroll
  for (int kb = 0; kb < D_; kb += 4) {
    if ((kb & 63) == 0)  // global_prefetch_b8 ahead of the k stream
      __builtin_prefetch(krow + kb + 64, 0, 3);
    v2f a;
    if (m < K_) { a = ld2(qrow + kb); } else { a.x = 0.0f; a.y = 0.0f; }
    acc = wmma4(a, ld2(krow + kb), acc);
  }
  if (half == 0) {  // only real slot rows 0..7
    const float scale = 0.08838834764831845f;  // 128^-0.5
#pragma unroll
    for (int j = 0; j < K_; ++j)
      dots[(size_t)b * K_ * N_ + (size_t)j * N_ + nt * 16 + m] = acc[j] * scale;
  }
}

// ------ updates[b,k,e] = sum_n attn_n[b,k,n] * v[b,n,e]  (M=8 padded) -------
__global__ __launch_bounds__(32)
void upd_av(const float* __restrict__ attn_n, const float* __restrict__ vbuf,
            float* __restrict__ upd) {
  const int lane = threadIdx.x;
  const int m = lane & 15, half = lane >> 4;
  const int et = blockIdx.x, b = blockIdx.y;
  const float* arow = attn_n + (size_t)b * K_ * N_ + (size_t)m * N_ + half * 2;
  const float* vb = vbuf + (size_t)b * N_ * D_ + et * 16 + m;
  v8f acc = {};
  for (int kb = 0; kb < N_; kb += 4) {
    if ((kb & 63) == 0)  // global_prefetch_b8 ahead of the v stream
      __builtin_prefetch(vb + (size_t)(kb + 64) * D_, 0, 3);
    v2f a;
    if (m < K_) { a = ld2(arow + kb); } else { a.x = 0.0f; a.y = 0.0f; }
    const int kr = kb + half * 2;
    v2f bf;
    bf.x = vb[(size_t)kr * D_];
    bf.y = vb[(size_t)(kr + 1) * D_];
    acc = wmma4(a, bf, acc);
  }
  if (half == 0) {
#pragma unroll
    for (int j = 0; j < K_; ++j)
      upd[(size_t)b * K_ * D_ + (size_t)j * D_ + et * 16 + m] = acc[j];
  }
}

// ---------------- elementwise kernels ---------------------------------------
__global__ __launch_bounds__(256)
void init_slots(const float* __restrict__ mu, const float* __restrict__ ls,
                const float* __restrict__ noise, float* __restrict__ slots) {
  int i = blockIdx.x * 256 + threadIdx.x;
  if (i >= B_ * K_ * D_) return;
  int d = i & (D_ - 1);
  slots[i] = mu[d] + expf(ls[d]) * noise[i];
}

__global__ __launch_bounds__(256)
void softmax_slots(const float* __restrict__ dots, float* __restrict__ attn) {
  int i = blockIdx.x * 256 + threadIdx.x;
  if (i >= B_ * N_) return;
  int b = i >> 12, n = i & (N_ - 1);
  const float* dp = dots + (size_t)b * K_ * N_ + n;
  float d[K_], mx = -3.402823466e+38f;
#pragma unroll
  for (int s = 0; s < K_; ++s) { d[s] = dp[(size_t)s * N_]; mx = fmaxf(mx, d[s]); }
  float sum = 0.0f;
#pragma unroll
  for (int s = 0; s < K_; ++s) { d[s] = expf(d[s] - mx); sum += d[s]; }
  float inv = 1.0f / sum;
  float* ap = attn + (size_t)b * K_ * N_ + n;
#pragma unroll
  for (int s = 0; s < K_; ++s) ap[(size_t)s * N_] = d[s] * inv;
}

__global__ __launch_bounds__(256)
void attn_rownorm(const float* __restrict__ attn, float* __restrict__ attn_n) {
  __shared__ float red[256];
  int row = blockIdx.x;  // (b*K + k), 512 rows
  const float* ap = attn + (size_t)row * N_;
  float s = 0.0f;
  for (int j = threadIdx.x; j < N_; j += 256) s += ap[j];
  red[threadIdx.x] = s;
  __syncthreads();
  for (int off = 128; off; off >>= 1) {
    if (threadIdx.x < off) red[threadIdx.x] += red[threadIdx.x + off];
    __syncthreads();
  }
  float inv = 1.0f / (red[0] + 1e-8f);
  float* op = attn_n + (size_t)row * N_;
  for (int j = threadIdx.x; j < N_; j += 256) op[j] = ap[j] * inv;
}

__device__ __forceinline__ float sigmoidf_(float x) {
  return 1.0f / (1.0f + expf(-x));
}

__global__ __launch_bounds__(256)
void gru_combine(const float* __restrict__ gx, const float* __restrict__ gh,
                 float* __restrict__ slots) {
  int i = blockIdx.x * 256 + threadIdx.x;
  if (i >= B_ * K_ * D_) return;
  int row = i >> 7, d = i & (D_ - 1);
  const float* gxr = gx + (size_t)row * (3 * D_);
  const float* ghr = gh + (size_t)row * (3 * D_);
  float r = sigmoidf_(gxr[d] + ghr[d]);
  float z = sigmoidf_(gxr[D_ + d] + ghr[D_ + d]);
  float n = tanhf(gxr[2 * D_ + d] + r * ghr[2 * D_ + d]);
  float h = slots[i];
  slots[i] = (1.0f - z) * n + z * h;
}

// ---------------------------------------------------------------------------
extern "C" void kernel_launch(void* const* d_in, const int* in_sizes, int n_in,
                              void* d_out, int out_size, void* d_ws,
                              size_t ws_size, hipStream_t stream) {
  const float* inputs   = (const float*)d_in[0];
  const float* noise    = (const float*)d_in[1];
  const float* ln_in_g  = (const float*)d_in[2];
  const float* ln_in_b  = (const float*)d_in[3];
  const float* ln_sl_g  = (const float*)d_in[4];
  const float* ln_sl_b  = (const float*)d_in[5];
  const float* ln_mlp_g = (const float*)d_in[6];
  const float* ln_mlp_b = (const float*)d_in[7];
  const float* slots_mu = (const float*)d_in[8];
  const float* slots_ls = (const float*)d_in[9];
  const float* Wq       = (const float*)d_in[10];
  const float* Wk       = (const float*)d_in[11];
  const float* Wv       = (const float*)d_in[12];
  const float* gru_Wih  = (const float*)d_in[13];
  const float* gru_Whh  = (const float*)d_in[14];
  const float* gru_bih  = (const float*)d_in[15];
  const float* gru_bhh  = (const float*)d_in[16];
  const float* mlp_W1   = (const float*)d_in[17];
  const float* mlp_b1   = (const float*)d_in[18];
  const float* mlp_W2   = (const float*)d_in[19];
  const float* mlp_b2   = (const float*)d_in[20];

  float* ws = (float*)d_ws;
  const size_t XN = (size_t)B_ * N_ * D_;  // 33,554,432 floats
  float* k_buf = ws;
  float* v_buf = ws + XN;
  float* small = v_buf + XN;
  float* dots   = small;
  float* attn   = dots + (size_t)B_ * K_ * N_;
  float* attn_n = attn + (size_t)B_ * K_ * N_;
  float* slots  = attn_n + (size_t)B_ * K_ * N_;
  float* s_n    = slots + B_ * K_ * D_;
  float* qb     = s_n + B_ * K_ * D_;
  float* upd    = qb + B_ * K_ * D_;
  float* gx     = upd + B_ * K_ * D_;
  float* gh     = gx + B_ * K_ * 3 * D_;
  float* h_n    = gh + B_ * K_ * 3 * D_;
  float* h1     = h_n + B_ * K_ * D_;

  const int R_in = B_ * N_;   // 262144 rows
  const int R_sl = B_ * K_;   // 512 rows

  // Fused LN + k/v projection (reads inputs once, writes k and v once).
  ln_kv_fused<<<R_in / 16, 256, 0, stream>>>(inputs, ln_in_g, ln_in_b, Wk, Wv,
                                             k_buf, v_buf);

  init_slots<<<(B_ * K_ * D_) / 256, 256, 0, stream>>>(slots_mu, slots_ls,
                                                       noise, slots);

  for (int it = 0; it < 3; ++it) {
    ln_rows_128<<<R_sl / 8, 256, 0, stream>>>(slots, ln_sl_g, ln_sl_b, s_n, R_sl);
    gemm_xwt<<<dim3(R_sl / 16, D_ / 16), 32, 0, stream>>>(
        s_n, Wq, nullptr, nullptr, qb, D_, D_, 0);
    dots_qk<<<dim3(N_ / 16, B_), 32, 0, stream>>>(qb, k_buf, dots);
    softmax_slots<<<(B_ * N_) / 256, 256, 0, stream>>>(dots, attn);
    attn_rownorm<<<B_ * K_, 256, 0, stream>>>(attn, attn_n);
    upd_av<<<dim3(D_ / 16, B_), 32, 0, stream>>>(attn_n, v_buf, upd);
    // GRU: gx = upd Wih^T + bih ; gh = slots Whh^T + bhh ; combine in place
    gemm_xwt<<<dim3(R_sl / 16, (3 * D_) / 16), 32, 0, stream>>>(
        upd, gru_Wih, gru_bih, nullptr, gx, D_, 3 * D_, 0);
    gemm_xwt<<<dim3(R_sl / 16, (3 * D_) / 16), 32, 0, stream>>>(
        slots, gru_Whh, gru_bhh, nullptr, gh, D_, 3 * D_, 0);
    gru_combine<<<(B_ * K_ * D_) / 256, 256, 0, stream>>>(gx, gh, slots);
    // MLP residual: slots += W2 gelu(W1 LN(slots) + b1) + b2
    ln_rows_128<<<R_sl / 8, 256, 0, stream>>>(slots, ln_mlp_g, ln_mlp_b, h_n, R_sl);
    gemm_xwt<<<dim3(R_sl / 16, MLPH_ / 16), 32, 0, stream>>>(
        h_n, mlp_W1, mlp_b1, nullptr, h1, D_, MLPH_, 1);
    gemm_xwt<<<dim3(R_sl / 16, D_ / 16), 32, 0, stream>>>(
        h1, mlp_W2, mlp_b2, slots, slots, MLPH_, D_, 0);
  }

  // outputs: slots (B,K,D) then attn (B,K,N), concatenated
  hipMemcpyAsync(d_out, slots, (size_t)B_ * K_ * D_ * sizeof(float),
                 hipMemcpyDeviceToDevice, stream);
  hipMemcpyAsync((float*)d_out + B_ * K_ * D_, attn,
                 (size_t)B_ * K_ * N_ * sizeof(float),
                 hipMemcpyDeviceToDevice, stream);
}